// GraphSAGE_31310311587980
// MI455X (gfx1250) — compile-verified
//
#include <hip/hip_runtime.h>
#include <hip/hip_bf16.h>

typedef __attribute__((ext_vector_type(2))) float v2f;
typedef __attribute__((ext_vector_type(8))) float v8f;

#define FEAT 128   // IN_FEATS == HID == 128 for this problem

// ---------------------------------------------------------------- utilities
__global__ void zero_f32_kernel(float* __restrict__ p, long long n) {
  long long i = (long long)blockIdx.x * blockDim.x + threadIdx.x;
  long long stride = (long long)gridDim.x * blockDim.x;
  for (; i < n; i += stride) p[i] = 0.0f;
}

__global__ void deg_count_kernel(const int* __restrict__ dst,
                                 float* __restrict__ deg, int nE) {
  int e = blockIdx.x * blockDim.x + threadIdx.x;
  if (e < nE) atomicAdd(&deg[dst[e]], 1.0f);
}

__global__ void invdeg_kernel(float* __restrict__ deg, int n) {
  int i = blockIdx.x * blockDim.x + threadIdx.x;
  if (i < n) deg[i] = 1.0f / fmaxf(deg[i], 1.0f);
}

// One edge handled by 32 threads; each thread moves 4 consecutive floats.
// Gathers hit L2 (feature matrix is L2-resident); scatter uses f32 atomics.
__global__ void scatter_add_kernel(const float* __restrict__ x,
                                   const int* __restrict__ src,
                                   const int* __restrict__ dst,
                                   float* __restrict__ agg, int nE) {
  long long t = (long long)blockIdx.x * blockDim.x + threadIdx.x;
  int e = (int)(t >> 5);
  if (e >= nE) return;
  int c = ((int)t & 31) << 2;               // chunk of 4 floats, 32 chunks = 128
  int s = src[e], d = dst[e];
  const float4 v = *reinterpret_cast<const float4*>(x + (size_t)s * FEAT + c);
  float* o = agg + (size_t)d * FEAT + c;
  atomicAdd(o + 0, v.x);
  atomicAdd(o + 1, v.y);
  atomicAdd(o + 2, v.z);
  atomicAdd(o + 3, v.w);
}

// ------------------------------------------------- fused SAGE layer (WMMA)
// OUT[m,n] = act( X[m,:]@Wself[:,n] + (AGG[m,:]*invd[m])@Wneigh[:,n] + bias[n] )
// One wave32 computes one 16x16 tile with V_WMMA_F32_16X16X4_F32 (exact f32).
__global__ __launch_bounds__(128)
void sage_gemm_wmma(const float* __restrict__ X,      // [M,K]
                    const float* __restrict__ AGG,    // [M,K] neighbor sums
                    const float* __restrict__ INVD,   // [M]   1/max(deg,1)
                    const float* __restrict__ Wself,  // [K,N]
                    const float* __restrict__ Wneigh, // [K,N]
                    const float* __restrict__ bias,   // [N]
                    float* __restrict__ OUT,          // [M,N]
                    int M, int N, int K, int relu) {
  const int lane = threadIdx.x & 31;
  const int wave = threadIdx.x >> 5;
  const int ntn  = N >> 4;
  const int tile = blockIdx.x * 4 + wave;
  const int total = ((M + 15) >> 4) * ntn;
  if (tile >= total) return;                 // wave-uniform: EXEC stays all-1s

  const int tm = tile / ntn;
  const int tn = tile - tm * ntn;
  const int m0 = tm << 4;
  const int n0 = tn << 4;

  const int half = lane >> 4;                // 0: lanes 0-15 (K=0,1)  1: K=2,3
  const int l15  = lane & 15;
  int arow = m0 + l15;                       // A-matrix row owned by this lane
  if (arow >= M) arow = M - 1;               // clamp loads only (EXEC unchanged)
  const int bcol = n0 + l15;                 // B-matrix column owned by this lane
  const float id = INVD[arow];

  const float* __restrict__ xrow = X   + (size_t)arow * K;
  const float* __restrict__ grow = AGG + (size_t)arow * K;

  v8f acc = {};
#pragma unroll 4
  for (int k0 = 0; k0 < K; k0 += 4) {
    const int ka = k0 + half * 2;            // this lane's first K index
    v2f a_s, a_g, b_s, b_n;
    // A fragments (ISA 7.12.2: 16x4 f32 A, lanes 0-15 K={0,1}, 16-31 K={2,3})
    a_s.x = xrow[ka];          a_s.y = xrow[ka + 1];
    a_g.x = grow[ka] * id;     a_g.y = grow[ka + 1] * id;   // fused mean
    // B fragments (4x16, row-striped across lanes, mirrored K split)
    b_s.x = Wself [(size_t)ka       * N + bcol];
    b_s.y = Wself [(size_t)(ka + 1) * N + bcol];
    b_n.x = Wneigh[(size_t)ka       * N + bcol];
    b_n.y = Wneigh[(size_t)(ka + 1) * N + bcol];
    acc = __builtin_amdgcn_wmma_f32_16x16x4_f32(false, a_s, false, b_s,
                                                (short)0, acc, false, false);
    acc = __builtin_amdgcn_wmma_f32_16x16x4_f32(false, a_g, false, b_n,
                                                (short)0, acc, false, false);
  }

  // Epilogue: C/D layout — VGPR r holds (M=m0+r, N=bcol) for lanes 0-15 and
  // (M=m0+r+8, N=bcol) for lanes 16-31.
  const float bv = bias[bcol];
  const int row0 = m0 + half * 8;
#pragma unroll
  for (int r = 0; r < 8; ++r) {
    const int row = row0 + r;
    float v = acc[r] + bv;
    if (relu) v = fmaxf(v, 0.0f);
    if (row < M) OUT[(size_t)row * N + bcol] = v;
  }
}

// ---------------------------------------------------------------- launcher
extern "C" void kernel_launch(void* const* d_in, const int* in_sizes, int n_in,
                              void* d_out, int out_size, void* d_ws, size_t ws_size,
                              hipStream_t stream) {
  const float* features = (const float*)d_in[0];
  const int*   src      = (const int*)  d_in[1];
  const int*   dst      = (const int*)  d_in[2];
  const float* Ws0      = (const float*)d_in[3];
  const float* Wn0      = (const float*)d_in[4];
  const float* b0       = (const float*)d_in[5];
  const float* Ws1      = (const float*)d_in[6];
  const float* Wn1      = (const float*)d_in[7];
  const float* b1       = (const float*)d_in[8];
  float* out = (float*)d_out;

  const int nNodes = in_sizes[0] / FEAT;     // 100000
  const int nE     = in_sizes[1];            // 1600000
  const int HID    = in_sizes[5];            // 128
  const int OUTF   = in_sizes[8];            // 64

  // Workspace: [deg/invdeg | agg sums (N x HID) | hidden h (N x HID)]
  float* ws  = (float*)d_ws;
  const size_t padN = ((size_t)nNodes + 127) & ~(size_t)127;
  float* deg = ws;
  float* agg = ws + padN;
  float* h   = agg + (size_t)nNodes * HID;

  const long long aggN   = (long long)nNodes * HID;
  const long long sth    = (long long)nE * 32;
  const int sblocks      = (int)((sth + 255) / 256);
  const int eblocks      = (nE + 255) / 256;
  const int nblocks      = (nNodes + 255) / 256;

  // ---- layer 0 ----
  zero_f32_kernel<<<4096, 256, 0, stream>>>(deg, (long long)padN + aggN);
  deg_count_kernel<<<eblocks, 256, 0, stream>>>(dst, deg, nE);
  scatter_add_kernel<<<sblocks, 256, 0, stream>>>(features, src, dst, agg, nE);
  invdeg_kernel<<<nblocks, 256, 0, stream>>>(deg, nNodes);

  const int tiles0 = ((nNodes + 15) / 16) * (HID / 16);
  sage_gemm_wmma<<<(tiles0 + 3) / 4, 128, 0, stream>>>(
      features, agg, deg, Ws0, Wn0, b0, h, nNodes, HID, FEAT, 1);

  // ---- layer 1 (reuse agg buffer and invdeg) ----
  zero_f32_kernel<<<4096, 256, 0, stream>>>(agg, aggN);
  scatter_add_kernel<<<sblocks, 256, 0, stream>>>(h, src, dst, agg, nE);

  const int tiles1 = ((nNodes + 15) / 16) * (OUTF / 16);
  sage_gemm_wmma<<<(tiles1 + 3) / 4, 128, 0, stream>>>(
      h, agg, deg, Ws1, Wn1, b1, out, nNodes, OUTF, HID, 0);
}